// ContinuousAlphagrid_63591285784878
// MI455X (gfx1250) — compile-verified
//
#include <hip/hip_runtime.h>
#include <cstdint>
#include <cstddef>

// ---- problem constants (from reference) ----
#define GRID_N      128
#define NEAR_T      0.2f
#define FAR_T       6.0f
#define STEP_SZ     0.003383f
#define FOCAL_F     1000.0f
#define N_RAYS      16384
#define N_SAMPLES   512
#define SPT         4                 // samples per thread (one b128 per output stream)
#define THREADS_PER_RAY (N_SAMPLES / SPT)   // 128 -> wave-uniform ray index

typedef float v4f __attribute__((ext_vector_type(4)));

__device__ __forceinline__ uint32_t expand_bits(uint32_t v) {
    v = (v * 65537u) & 0xFF0000FFu;
    v = (v * 257u)   & 0x0F00F00Fu;
    v = (v * 17u)    & 0xC30C30C3u;
    v = (v * 5u)     & 0x49249249u;
    return v;
}

__global__ __launch_bounds__(256)
void alphagrid_sample_kernel(const float* __restrict__ rays,
                             const float* __restrict__ aabb,
                             const float* __restrict__ jitter,
                             const int*   __restrict__ bitfield,
                             float*       __restrict__ out)
{
    const int tid = blockIdx.x * blockDim.x + threadIdx.x;
    const int ray = tid >> 7;                 // uniform across a wave32 (32 | 128)
    const int s0  = (tid & (THREADS_PER_RAY - 1)) << 2;

    // Output sections (concatenated flat in reference return order)
    float* __restrict__ o_xyz = out;                                        // (R,S,4)
    float* __restrict__ o_rv  = out + (size_t)N_RAYS * N_SAMPLES * 4;       // (R,S)
    float* __restrict__ o_zv  = o_rv + (size_t)N_RAYS * N_SAMPLES;          // (R,S)
    float* __restrict__ o_di  = o_zv + (size_t)N_RAYS * N_SAMPLES;          // (R,S)
    float* __restrict__ o_wv  = o_di + (size_t)N_RAYS * N_SAMPLES;          // (R,)

    if (tid < N_RAYS) o_wv[tid] = 1.0f;       // whole_valid = ones

    // AABB (tiny, cache resident)
    const float lox = aabb[0], loy = aabb[1], loz = aabb[2];
    const float hix = aabb[3], hiy = aabb[4], hiz = aabb[5];

    // Ray origin / direction (wave-uniform -> L1 broadcast hit)
    const float ox = rays[ray * 6 + 0], oy = rays[ray * 6 + 1], oz = rays[ray * 6 + 2];
    const float dx = rays[ray * 6 + 3], dy = rays[ray * 6 + 4], dz = rays[ray * 6 + 5];

    // t_min per reference: clip(max(min(rate_a, rate_b)), NEAR, FAR)
    const float vx = (dx == 0.0f) ? 1e-6f : dx;
    const float vy = (dy == 0.0f) ? 1e-6f : dy;
    const float vz = (dz == 0.0f) ? 1e-6f : dz;
    const float tmx = fminf((hix - ox) / vx, (lox - ox) / vx);
    const float tmy = fminf((hiy - oy) / vy, (loy - oy) / vy);
    const float tmz = fminf((hiz - oz) / vz, (loz - oz) / vz);
    const float tmin = fminf(fmaxf(fmaxf(fmaxf(tmx, tmy), tmz), NEAR_T), FAR_T);

    // Jitter: streamed once -> non-temporal b128 load; prefetch ahead (gfx1250 path)
    const size_t jbase = (size_t)ray * N_SAMPLES + (size_t)s0;
    __builtin_prefetch(jitter + jbase + 4096, 0, 0);
    const v4f j4 = __builtin_nontemporal_load((const v4f*)(jitter + jbase));
    const float jnext = (s0 < N_SAMPLES - SPT) ? jitter[jbase + SPT] : 0.0f;

    float jit[SPT + 1] = { j4.x, j4.y, j4.z, j4.w, jnext };
    float zv[SPT + 1];
#pragma unroll
    for (int j = 0; j < SPT + 1; ++j)
        zv[j] = tmin + STEP_SZ * ((float)(s0 + j) + jit[j]);

    v4f zvec, dvec, rvec;
    v4f xyzv[SPT];

#pragma unroll
    for (int j = 0; j < SPT; ++j) {
        const float z  = zv[j];
        const float px = fmaf(dx, z, ox);
        const float py = fmaf(dy, z, oy);
        const float pz = fmaf(dz, z, oz);

        // mask_out = any(lo > p || p > hi)
        const bool inb = !((lox > px) || (px > hix) ||
                           (loy > py) || (py > hiy) ||
                           (loz > pz) || (pz > hiz));

        // cascade: CASCADE=2, BOUND=2 -> bound = (max|p| >= 1) ? 2 : 1
        const float mx    = fmaxf(fabsf(px), fmaxf(fabsf(py), fabsf(pz)));
        const float bound = (mx >= 1.0f) ? 2.0f : 1.0f;
        const float denom = bound - bound * (1.0f / (float)GRID_N);  // bound - half_grid

        const float cx = fminf(fmaxf(px / denom, -1.0f), 1.0f);
        const float cy = fminf(fmaxf(py / denom, -1.0f), 1.0f);
        const float cz = fminf(fmaxf(pz / denom, -1.0f), 1.0f);
        const uint32_t ix = (uint32_t)(((cx + 1.0f) * 0.5f) * (float)(GRID_N - 1));
        const uint32_t iy = (uint32_t)(((cy + 1.0f) * 0.5f) * (float)(GRID_N - 1));
        const uint32_t iz = (uint32_t)(((cz + 1.0f) * 0.5f) * (float)(GRID_N - 1));

        const uint32_t idx = expand_bits(ix) | (expand_bits(iy) << 1) | (expand_bits(iz) << 2);
        const int bits = bitfield[idx >> 3];                 // 2 MB table: L2-resident
        const bool alpha = ((bits >> (idx & 7)) & 1) != 0;

        const bool  valid = inb && alpha;
        const float fv    = valid ? 1.0f : 0.0f;

        zvec[j] = z;
        rvec[j] = fv;
        dvec[j] = ((s0 + j) == (N_SAMPLES - 1)) ? 0.0f : (zv[j + 1] - z);

        v4f xm;
        xm.x = px * fv;
        xm.y = py * fv;
        xm.z = pz * fv;
        xm.w = (z / FOCAL_F) * fv;
        xyzv[j] = xm;
    }

    // Streaming outputs: non-temporal b128 stores (don't thrash L2; bitfield stays hot)
    const size_t vb = (size_t)tid;
    __builtin_nontemporal_store(zvec, (v4f*)o_zv + vb);
    __builtin_nontemporal_store(dvec, (v4f*)o_di + vb);
    __builtin_nontemporal_store(rvec, (v4f*)o_rv + vb);
    v4f* xp = (v4f*)o_xyz + vb * SPT;
#pragma unroll
    for (int j = 0; j < SPT; ++j)
        __builtin_nontemporal_store(xyzv[j], xp + j);
}

extern "C" void kernel_launch(void* const* d_in, const int* in_sizes, int n_in,
                              void* d_out, int out_size, void* d_ws, size_t ws_size,
                              hipStream_t stream) {
    (void)in_sizes; (void)n_in; (void)out_size; (void)d_ws; (void)ws_size;
    const float* rays     = (const float*)d_in[0];   // (16384, 6)
    const float* aabb     = (const float*)d_in[1];   // (2, 3)
    const float* jitter   = (const float*)d_in[2];   // (16384, 512)
    const int*   bitfield = (const int*)  d_in[3];   // (524288,)
    float* out = (float*)d_out;

    const int total_threads = (N_RAYS * N_SAMPLES) / SPT;   // 2,097,152
    const int block = 256;                                   // 8 wave32s
    const int grid  = total_threads / block;                 // 8192 blocks

    alphagrid_sample_kernel<<<grid, block, 0, stream>>>(rays, aabb, jitter, bitfield, out);
}